// GATSingleAttentionHead_7164005450397
// MI455X (gfx1250) — compile-verified
//
#include <hip/hip_runtime.h>
#include <hip/hip_bf16.h>
#include <math.h>

// GAT single attention head, MI455X (gfx1250, wave32).
// Pipeline (all on `stream`, ordered by stream semantics):
//   K0 init      : zero d_out (h_new accumulator), init m_raw/denom
//   K1 gemm_wmma : Wh = x @ W.T via v_wmma_f32_16x16x4_f32 (fp32-exact)
//   K2 score     : s_src[n] = Wh[n]·a_src, s_dst[n] = Wh[n]·a_dst
//   K3 edge max  : m[dst] = segment_max(leaky_relu(s_src[src]+s_dst[dst]))
//   K4 edge sum  : denom[dst] += exp(e - m[dst])
//   K5 scatter   : d_out[dst] += (exp(e-m)/denom) * Wh[src]   (wave per edge)
//   K6 finalize  : d_out = elu(d_out + bias)

#define IN_F   512
#define OUT_F  128

typedef __attribute__((ext_vector_type(2))) float v2f;
typedef __attribute__((ext_vector_type(8))) float v8f;

// ---- order-preserving float <-> uint mapping (for atomic segment-max) ----
__device__ __forceinline__ unsigned ord_enc(float f) {
    unsigned u = __float_as_uint(f);
    return (u & 0x80000000u) ? ~u : (u | 0x80000000u);
}
__device__ __forceinline__ float ord_dec(unsigned m) {
    unsigned u = (m & 0x80000000u) ? (m & 0x7fffffffu) : ~m;
    return __uint_as_float(u);
}
__device__ __forceinline__ float lrelu(float v) { return v > 0.0f ? v : 0.2f * v; }

// ---- K0: init accumulators -------------------------------------------------
__global__ void init_kernel(float* __restrict__ out, unsigned* __restrict__ m_raw,
                            float* __restrict__ denom, int n_nodes) {
    int i = blockIdx.x * blockDim.x + threadIdx.x;
    if (i < n_nodes * OUT_F) out[i] = 0.0f;
    if (i < n_nodes) { m_raw[i] = 0u; denom[i] = 0.0f; }
}

// ---- K1: Wh = x @ W.T via f32 WMMA ----------------------------------------
// One wave per 16x16 tile of Wh. mtile = blockIdx.x (625), ntile = wave (8).
// 32-bit A 16x4 fragment layout (ISA 7.12.2): lane L holds row M=L%16,
// K = k0 + 2*(L/16) + {0,1} in 2 VGPRs.  B 4x16 mirrors with N=L%16.
// Since B[k][n] = W[n][k], both fragments are consecutive-pair loads along
// the K-contiguous rows of x and W -> 8B-aligned float2 loads.
__global__ void __launch_bounds__(256)
gemm_wmma_kernel(const float* __restrict__ x, const float* __restrict__ W,
                 float* __restrict__ Wh) {
    const int ntile = threadIdx.x >> 5;     // wave in block: 0..7 -> N tile
    const int lane  = threadIdx.x & 31;
    const int mtile = blockIdx.x;           // 0..624
    const int row   = lane & 15;
    const int khalf = lane >> 4;            // 0 or 1 -> K sub-pair

    const float* __restrict__ arow = x + (size_t)(mtile * 16 + row) * IN_F + 2 * khalf;
    const float* __restrict__ brow = W + (size_t)(ntile * 16 + row) * IN_F + 2 * khalf;

    v8f acc = {};
#pragma unroll 8
    for (int k0 = 0; k0 < IN_F; k0 += 4) {
        v2f a = *(const v2f*)(arow + k0);
        v2f b = *(const v2f*)(brow + k0);
        // 8 args: (neg_a, A, neg_b, B, c_mod, C, reuse_a, reuse_b)
        acc = __builtin_amdgcn_wmma_f32_16x16x4_f32(
            false, a, false, b, (short)0, acc, false, false);
    }
    // C/D layout: lane L, VGPR v -> M = (L/16)*8 + v, N = L%16
    float* __restrict__ orow = Wh + (size_t)(mtile * 16 + khalf * 8) * OUT_F
                                  + ntile * 16 + row;
#pragma unroll
    for (int v = 0; v < 8; ++v) orow[(size_t)v * OUT_F] = acc[v];
}

// ---- K2: per-node attention scores ----------------------------------------
__global__ void score_kernel(const float* __restrict__ Wh,
                             const float* __restrict__ attn_w,
                             float* __restrict__ s_src, float* __restrict__ s_dst,
                             int n_nodes) {
    int i = blockIdx.x * blockDim.x + threadIdx.x;
    if (i >= n_nodes) return;
    const float* __restrict__ rowp = Wh + (size_t)i * OUT_F;
    float a = 0.0f, b = 0.0f;
#pragma unroll 8
    for (int f = 0; f < OUT_F; ++f) {
        float w = rowp[f];
        a = fmaf(w, attn_w[f], a);
        b = fmaf(w, attn_w[OUT_F + f], b);
    }
    s_src[i] = a;
    s_dst[i] = b;
}

// ---- K3: segment max over edges (atomic max on ordered encoding) ----------
__global__ void edge_max_kernel(const int* __restrict__ src, const int* __restrict__ dst,
                                const float* __restrict__ s_src,
                                const float* __restrict__ s_dst,
                                unsigned* __restrict__ m_raw, int n_edges) {
    int i = blockIdx.x * blockDim.x + threadIdx.x;
    if (i >= n_edges) return;
    float e = lrelu(s_src[src[i]] + s_dst[dst[i]]);
    atomicMax(&m_raw[dst[i]], ord_enc(e));
}

// ---- K4: softmax denominator ----------------------------------------------
__global__ void edge_sum_kernel(const int* __restrict__ src, const int* __restrict__ dst,
                                const float* __restrict__ s_src,
                                const float* __restrict__ s_dst,
                                const unsigned* __restrict__ m_raw,
                                float* __restrict__ denom, int n_edges) {
    int i = blockIdx.x * blockDim.x + threadIdx.x;
    if (i >= n_edges) return;
    int d = dst[i];
    float e = lrelu(s_src[src[i]] + s_dst[d]);
    float m = ord_dec(m_raw[d]);            // finite: this dst has >=1 edge
    atomicAdd(&denom[d], __expf(e - m));
}

// ---- K5: weighted feature scatter (one wave per edge) ---------------------
// Lane l covers features {l, l+32, l+64, l+96}: each 32-lane atomic burst is
// a contiguous 128B slice -> coalesced L2 atomics (Wh & out are L2-resident).
__global__ void __launch_bounds__(256)
scatter_kernel(const int* __restrict__ src, const int* __restrict__ dst,
               const float* __restrict__ Wh, const float* __restrict__ s_src,
               const float* __restrict__ s_dst, const unsigned* __restrict__ m_raw,
               const float* __restrict__ denom, float* __restrict__ out, int n_edges) {
    int edge = blockIdx.x * 8 + (threadIdx.x >> 5);
    int lane = threadIdx.x & 31;
    if (edge >= n_edges) return;
    int s = src[edge], d = dst[edge];
    float e = lrelu(s_src[s] + s_dst[d]);
    float alpha = __expf(e - ord_dec(m_raw[d])) / denom[d];
    const float* __restrict__ wrow = Wh + (size_t)s * OUT_F;
    float* __restrict__ orow = out + (size_t)d * OUT_F;
#pragma unroll
    for (int f = lane; f < OUT_F; f += 32)
        atomicAdd(&orow[f], alpha * wrow[f]);
}

// ---- K6: out = elu(h_new + bias) ------------------------------------------
__global__ void finalize_kernel(float* __restrict__ out, const float* __restrict__ bias,
                                int total) {
    int i = blockIdx.x * blockDim.x + threadIdx.x;
    if (i >= total) return;
    float v = out[i] + bias[i & (OUT_F - 1)];
    out[i] = v > 0.0f ? v : expm1f(v);
}

extern "C" void kernel_launch(void* const* d_in, const int* in_sizes, int n_in,
                              void* d_out, int out_size, void* d_ws, size_t ws_size,
                              hipStream_t stream) {
    const float* x      = (const float*)d_in[0];
    const int*   src    = (const int*)d_in[1];
    const int*   dst    = (const int*)d_in[2];
    const float* W      = (const float*)d_in[3];
    const float* attn_w = (const float*)d_in[4];
    const float* bias   = (const float*)d_in[5];
    float* out = (float*)d_out;

    const int n_nodes = in_sizes[0] / IN_F;   // 10000 (multiple of 16)
    const int n_edges = in_sizes[1];          // 640000

    // Workspace layout (floats): Wh | s_src | s_dst | m_raw(u32) | denom
    float*    Wh    = (float*)d_ws;
    float*    s_src = Wh + (size_t)n_nodes * OUT_F;
    float*    s_dst = s_src + n_nodes;
    unsigned* m_raw = (unsigned*)(s_dst + n_nodes);
    float*    denom = (float*)(m_raw + n_nodes);

    const int total = n_nodes * OUT_F;

    init_kernel<<<(total + 255) / 256, 256, 0, stream>>>(out, m_raw, denom, n_nodes);

    gemm_wmma_kernel<<<n_nodes / 16, 256, 0, stream>>>(x, W, Wh);

    score_kernel<<<(n_nodes + 255) / 256, 256, 0, stream>>>(Wh, attn_w, s_src, s_dst, n_nodes);

    edge_max_kernel<<<(n_edges + 255) / 256, 256, 0, stream>>>(src, dst, s_src, s_dst,
                                                               m_raw, n_edges);

    edge_sum_kernel<<<(n_edges + 255) / 256, 256, 0, stream>>>(src, dst, s_src, s_dst,
                                                               m_raw, denom, n_edges);

    scatter_kernel<<<(n_edges + 7) / 8, 256, 0, stream>>>(src, dst, Wh, s_src, s_dst,
                                                          m_raw, denom, out, n_edges);

    finalize_kernel<<<(total + 255) / 256, 256, 0, stream>>>(out, bias, total);
}